// BalancedCELoss_40870908788864
// MI455X (gfx1250) — compile-verified
//
#include <hip/hip_runtime.h>
#include <hip/hip_bf16.h>

typedef float v4f __attribute__((ext_vector_type(4)));
typedef int   v4i __attribute__((ext_vector_type(4)));

#define BLOCKS  4096
#define THREADS 256

// ln(2): log(x) = log2(x) * LN2 ; target==1 weight 2.0, else 1.0
#define LN2F 0.69314718055994530942f

__device__ __forceinline__ float wave32_sum(float v) {
    // full 32-lane xor tree (wave32 on gfx1250)
    #pragma unroll
    for (int off = 16; off > 0; off >>= 1)
        v += __shfl_xor(v, off, 32);
    return v;
}

__global__ __launch_bounds__(THREADS) void bce_partials_kernel(
    const float* __restrict__ p, const int* __restrict__ t,
    float* __restrict__ partials, int n) {

    const int tid    = blockIdx.x * blockDim.x + threadIdx.x;
    const int stride = gridDim.x * blockDim.x;
    const int n4     = n >> 2;

    const v4f* __restrict__ p4 = (const v4f*)p;
    const v4i* __restrict__ t4 = (const v4i*)t;

    float acc = 0.0f;

    for (int i = tid; i < n4; i += stride) {
        // prefetch two strides ahead (gfx1250 global_prefetch_b8 path)
        if (i + 2 * stride < n4) {
            __builtin_prefetch((const void*)(p4 + i + 2 * stride), 0, 1);
            __builtin_prefetch((const void*)(t4 + i + 2 * stride), 0, 1);
        }
        // streamed once, 268MB > 192MB L2 -> non-temporal loads
        v4f pv = __builtin_nontemporal_load(p4 + i);
        v4i tv = __builtin_nontemporal_load(t4 + i);
        #pragma unroll
        for (int k = 0; k < 4; ++k) {
            const bool one = (tv[k] != 0);
            const float x  = one ? pv[k] : (1.0f - pv[k]);
            const float w  = one ? (2.0f * LN2F) : LN2F;
            acc = fmaf(w, __log2f(x), acc);   // one v_log_f32 + one v_fma per element
        }
    }

    // scalar tail (N is a multiple of 4, but stay correct in general)
    for (int i = (n4 << 2) + tid; i < n; i += stride) {
        const int   tv = t[i];
        const float pv = p[i];
        const bool one = (tv != 0);
        const float x  = one ? pv : (1.0f - pv);
        const float w  = one ? (2.0f * LN2F) : LN2F;
        acc = fmaf(w, __log2f(x), acc);
    }

    // deterministic reduction: wave32 tree -> LDS across 8 waves -> block partial
    acc = wave32_sum(acc);

    __shared__ float smem[THREADS / 32];
    const int lane = threadIdx.x & 31;
    const int wid  = threadIdx.x >> 5;
    if (lane == 0) smem[wid] = acc;
    __syncthreads();

    if (wid == 0) {
        float v = (lane < (THREADS / 32)) ? smem[lane] : 0.0f;
        #pragma unroll
        for (int off = (THREADS / 64); off > 0; off >>= 1)
            v += __shfl_xor(v, off, 32);
        if (lane == 0) partials[blockIdx.x] = v;
    }
}

__global__ __launch_bounds__(THREADS) void bce_finalize_kernel(
    const float* __restrict__ partials, int nparts,
    float* __restrict__ out, float inv_n) {

    float acc = 0.0f;
    // fixed-order accumulation: deterministic across replays
    for (int i = threadIdx.x; i < nparts; i += THREADS)
        acc += partials[i];

    acc = wave32_sum(acc);

    __shared__ float smem[THREADS / 32];
    const int lane = threadIdx.x & 31;
    const int wid  = threadIdx.x >> 5;
    if (lane == 0) smem[wid] = acc;
    __syncthreads();

    if (wid == 0) {
        float v = (lane < (THREADS / 32)) ? smem[lane] : 0.0f;
        #pragma unroll
        for (int off = (THREADS / 64); off > 0; off >>= 1)
            v += __shfl_xor(v, off, 32);
        if (lane == 0) out[0] = -v * inv_n;   // loss = -mean(per_el)
    }
}

extern "C" void kernel_launch(void* const* d_in, const int* in_sizes, int n_in,
                              void* d_out, int out_size, void* d_ws, size_t ws_size,
                              hipStream_t stream) {
    const float* p = (const float*)d_in[0];
    const int*   t = (const int*)d_in[1];
    float*       o = (float*)d_out;
    float* partials = (float*)d_ws;           // 4096 floats = 16 KB scratch

    const int n = in_sizes[0];
    const float inv_n = 1.0f / (float)n;

    bce_partials_kernel<<<BLOCKS, THREADS, 0, stream>>>(p, t, partials, n);
    bce_finalize_kernel<<<1, THREADS, 0, stream>>>(partials, BLOCKS, o, inv_n);
}